// DistanceLoss_18227841204844
// MI455X (gfx1250) — compile-verified
//
#include <hip/hip_runtime.h>
#include <hip/hip_bf16.h>

typedef float v2f __attribute__((ext_vector_type(2)));
typedef float v8f __attribute__((ext_vector_type(8)));

namespace {
constexpr int BATCH = 4;
constexpr int HH = 512;
constexpr int WW = 512;
constexpr int KK = 64;
constexpr int HW = HH * WW;
constexpr float L_SEP = 300.0f;
constexpr float L_MEAN = 300.0f;

constexpr int BK = BATCH * KK;
// workspace layout (float offsets)
constexpr int WS_CNT   = 0;             // BK
constexpr int WS_SUM   = BK;            // 3*BK
constexpr int WS_MEAN  = 4 * BK;        // 3*BK  (raw means, used for inter d2)
constexpr int WS_MEANZ = 7 * BK;        // 3*BK  (bg-zeroed means, used for intra)
constexpr int WS_M2P1  = 10 * BK;       // BK    (|mean|^2 + 1)
constexpr int WS_COEF  = 11 * BK;       // BK    (3000/((HW-c)*sqrt(c)), 0 for bg)
constexpr int WS_ISC   = 12 * BK;       // BK    (cf/(3*cnt))
constexpr int WS_CT    = 13 * BK;       // BATCH
constexpr int WS_MSEP  = 13 * BK + BATCH;     // BATCH
constexpr int WS_ACC   = 13 * BK + 2 * BATCH; // BATCH
constexpr int WS_TOTAL = 13 * BK + 3 * BATCH;

constexpr int TPB = 256;                 // 8 waves (wave32)
constexpr int WAVES = TPB / 32;
constexpr int PIXBLK = 2048;             // pixels per block in main kernel
constexpr int GROUPS = PIXBLK / 16;      // 16-pixel WMMA groups per block
constexpr int BLOCKS_PER_B = HW / PIXBLK;      // 128
constexpr int CNT_BLOCKS_PER_B = 64;           // count kernel: 4096 px/block
}

__device__ __forceinline__ float huber1(float d) {
    float a = fabsf(d);
    return a < 1.0f ? 0.5f * a * a : a - 0.5f;
}

__global__ void k_zero(float* __restrict__ ws) {
    int i = blockIdx.x * blockDim.x + threadIdx.x;
    if (i < WS_TOTAL) ws[i] = 0.0f;
}

// Pass 1: per-(b,k) pixel counts and channel sums (LDS-accumulated, then global atomics)
__global__ __launch_bounds__(TPB) void k_count(const float* __restrict__ pred,
                                               const float* __restrict__ targ,
                                               const int* __restrict__ pal,
                                               float* __restrict__ ws) {
    __shared__ float cntL[KK];
    __shared__ float sumL[KK * 3];
    __shared__ int palL[KK];
    int tid = threadIdx.x;
    for (int i = tid; i < KK; i += TPB) { cntL[i] = 0.0f; palL[i] = pal[i]; }
    for (int i = tid; i < KK * 3; i += TPB) sumL[i] = 0.0f;
    __syncthreads();

    int b = blockIdx.x / CNT_BLOCKS_PER_B;
    int chunk = blockIdx.x % CNT_BLOCKS_PER_B;
    int pixPerBlock = HW / CNT_BLOCKS_PER_B;
    int base = chunk * pixPerBlock;
    const float* p0 = pred + (size_t)(b * 3 + 0) * HW;
    const float* p1 = pred + (size_t)(b * 3 + 1) * HW;
    const float* p2 = pred + (size_t)(b * 3 + 2) * HW;
    const float* t0 = targ + (size_t)(b * 3 + 0) * HW;
    const float* t1 = targ + (size_t)(b * 3 + 1) * HW;
    const float* t2 = targ + (size_t)(b * 3 + 2) * HW;

    for (int i = base + tid; i < base + pixPerBlock; i += TPB) {
        float pid = t0[i] * 65536.0f + t1[i] * 256.0f + t2[i];
        int ks = -1;
        int gu = (int)pid;
        if (gu >= 0 && gu < KK && (float)palL[gu] == pid) {
            ks = gu;                         // fast path: palette == arange
        } else {
            for (int j = 0; j < KK; ++j)
                if ((float)palL[j] == pid) { ks = j; break; }
        }
        if (ks >= 0) {
            atomicAdd(&cntL[ks], 1.0f);
            atomicAdd(&sumL[ks * 3 + 0], p0[i]);
            atomicAdd(&sumL[ks * 3 + 1], p1[i]);
            atomicAdd(&sumL[ks * 3 + 2], p2[i]);
        }
    }
    __syncthreads();
    float* cnt = ws + WS_CNT;
    float* sum = ws + WS_SUM;
    for (int k = tid; k < KK; k += TPB) {
        atomicAdd(&cnt[b * KK + k], cntL[k]);
        atomicAdd(&sum[(b * KK + k) * 3 + 0], sumL[k * 3 + 0]);
        atomicAdd(&sum[(b * KK + k) * 3 + 1], sumL[k * 3 + 1]);
        atomicAdd(&sum[(b * KK + k) * 3 + 2], sumL[k * 3 + 2]);
    }
}

// Pass 2: means, coefficients, ct[b], and K x K mean-repulsion term. One block of BK threads.
__global__ __launch_bounds__(BK) void k_prep(const unsigned char* __restrict__ no_bg,
                                             const float* __restrict__ dw,
                                             const int* __restrict__ pal,
                                             float* __restrict__ ws) {
    __shared__ float smx[BK], smy[BK], smz[BK], scf[BK];
    __shared__ float sPen[BK], sNp[BK];
    int i = threadIdx.x;               // i = b*KK + k
    int b = i / KK, k = i % KK;
    float c = ws[WS_CNT + i];
    float inv = c > 0.0f ? 1.0f / c : 0.0f;
    float mx = ws[WS_SUM + i * 3 + 0] * inv;
    float my = ws[WS_SUM + i * 3 + 1] * inv;
    float mz = ws[WS_SUM + i * 3 + 2] * inv;
    ws[WS_MEAN + i * 3 + 0] = mx;
    ws[WS_MEAN + i * 3 + 1] = my;
    ws[WS_MEAN + i * 3 + 2] = mz;
    bool isbg = (pal[k] == 0);
    float zx = isbg ? 0.0f : mx, zy = isbg ? 0.0f : my, zz = isbg ? 0.0f : mz;
    ws[WS_MEANZ + i * 3 + 0] = zx;
    ws[WS_MEANZ + i * 3 + 1] = zy;
    ws[WS_MEANZ + i * 3 + 2] = zz;
    ws[WS_M2P1 + i] = mx * mx + my * my + mz * mz + 1.0f;
    // JAX bool arrays are 1 byte/element
    bool nb = no_bg[b] != 0;
    bool counted = (!isbg) || (!nb);
    float cf = counted ? 1.0f : 0.0f;
    ws[WS_COEF + i] = (!isbg && c > 0.0f)
                          ? (L_SEP * 10.0f) / (sqrtf(c) * ((float)HW - c))
                          : 0.0f;
    ws[WS_ISC + i] = (counted && c > 0.0f) ? 1.0f / (3.0f * c) : 0.0f;
    smx[i] = zx; smy[i] = zy; smz[i] = zz; scf[i] = cf;
    __syncthreads();

    float pen = 0.0f, np = 0.0f;
    if (cf > 0.0f) {
        for (int j = k + 1; j < KK; ++j) {
            int jj = b * KK + j;
            if (scf[jj] > 0.0f) {
                float dx = zx - smx[jj], dy = zy - smy[jj], dz = zz - smz[jj];
                float sqd = dx * dx + dy * dy + dz * dz;
                pen += dw[k * KK + j] * L_MEAN / (sqd + 1.0f);
                np += 1.0f;
            }
        }
    }
    sPen[i] = pen; sNp[i] = np;
    __syncthreads();
    if (i < BATCH) {
        float tp = 0.0f, tn = 0.0f, tc = 0.0f;
        for (int kk = 0; kk < KK; ++kk) {
            tp += sPen[i * KK + kk];
            tn += sNp[i * KK + kk];
            tc += scf[i * KK + kk];
        }
        ws[WS_CT + i] = fmaxf(tc, 1.0f);
        ws[WS_MSEP + i] = tn > 0.0f ? tp / fmaxf(tn, 1.0f) : 0.0f;
    }
}

// Pass 3: main per-pixel loss via V_WMMA_F32_16X16X4_F32.
//   A = [-2*x0, -2*x1 | -2*x2, 1]   (16 pixels x 4, pad slot carries 1.0)
//   B = [m0, m1 | m2, |M|^2+1]      (4 x 16 instances)
//   C = |P|^2 per pixel (broadcast across lanes via VGPR layout)
//   => D[m][n] = 1 + ||P_m - M_n||^2 directly out of the WMMA.
// Inner loop per element: one ds_load (pre-masked weight), one v_rcp, one fma.
// dwT has a zeroed diagonal (other-mask) and an extra all-zero row KK (invalid pixels),
// so the gather needs no compares / EXEC manipulation.
__global__ __launch_bounds__(TPB) void k_main(const float* __restrict__ pred,
                                              const float* __restrict__ targ,
                                              const float* __restrict__ dw,
                                              const int* __restrict__ pal,
                                              const float* __restrict__ wsc,
                                              float* __restrict__ accum) {
    __shared__ float dwT[(KK + 1) * KK];  // dwT[j*K + n] = dw[n][j], diag=0, row KK = 0
    __shared__ float mzL[KK * 3];         // means_z for this b
    __shared__ float iscL[KK];
    __shared__ int palL[KK];
    __shared__ float p2s[WAVES][16];      // per-wave scratch: |P|^2 of the 16 pixels
    __shared__ int kss[WAVES][16];        // per-wave scratch: instance id of the 16 pixels
    __shared__ float red[TPB];

    int tid = threadIdx.x;
    int b = blockIdx.x / BLOCKS_PER_B;
    int chunk = blockIdx.x % BLOCKS_PER_B;
    int base = chunk * PIXBLK;

    for (int i = tid; i < (KK + 1) * KK; i += TPB) {
        int j = i / KK, n = i % KK;       // dwT[j][n] = dist_weights[n][j]
        float v = 0.0f;
        if (j < KK && j != n) v = dw[n * KK + j];
        dwT[i] = v;
    }
    for (int i = tid; i < KK; i += TPB) {
        palL[i] = pal[i];
        iscL[i] = wsc[WS_ISC + b * KK + i];
    }
    for (int i = tid; i < KK * 3; i += TPB) mzL[i] = wsc[WS_MEANZ + (b * KK) * 3 + i];
    __syncthreads();

    int wave = tid >> 5, lane = tid & 31, lid = lane & 15;
    bool hi = lane >= 16;

    // Loop-invariant per-lane data: 4 B-matrix tiles (instances n = t*16+lid) + coef.
    // B layout (4x16 f32): lanes 0-15 hold rows K={0,1} for column N=lane;
    // lanes 16-31 hold rows K={2,3}. Row 3 carries |M|^2 + 1.
    v2f bmt[4];
    float cfR[4];
#pragma unroll
    for (int t = 0; t < 4; ++t) {
        int idx = b * KK + t * 16 + lid;
        bmt[t].x = hi ? wsc[WS_MEAN + idx * 3 + 2] : wsc[WS_MEAN + idx * 3 + 0];
        bmt[t].y = hi ? wsc[WS_M2P1 + idx] : wsc[WS_MEAN + idx * 3 + 1];
        cfR[t] = wsc[WS_COEF + idx];
    }

    const float* p0 = pred + (size_t)(b * 3 + 0) * HW;
    const float* p1 = pred + (size_t)(b * 3 + 1) * HW;
    const float* p2 = pred + (size_t)(b * 3 + 2) * HW;
    const float* t0 = targ + (size_t)(b * 3 + 0) * HW;
    const float* t1 = targ + (size_t)(b * 3 + 1) * HW;
    const float* t2 = targ + (size_t)(b * 3 + 2) * HW;

    float acc = 0.0f;

    for (int g = wave; g < GROUPS; g += WAVES) {
        int p = base + g * 16;
        int pix = p + lid;
        float x0 = p0[pix];
        float x1 = p1[pix];
        float x2 = p2[pix];
        // A-matrix 16x4 f32: lanes 0-15 carry K={0,1} = (-2*ch0, -2*ch1),
        // lanes 16-31 carry K={2,3} = (-2*ch2, 1.0).
        v2f a;
        a.x = hi ? -2.0f * x2 : -2.0f * x0;
        a.y = hi ? 1.0f : -2.0f * x1;

        if (!hi) {  // lanes 0-15 each own pixel `lid`: pid lookup + intra huber term
            float pid = t0[pix] * 65536.0f + t1[pix] * 256.0f + t2[pix];
            int ks = -1;
            int gu = (int)pid;
            if (gu >= 0 && gu < KK && (float)palL[gu] == pid) {
                ks = gu;
            } else {
                for (int j = 0; j < KK; ++j)
                    if ((float)palL[j] == pid) { ks = j; break; }
            }
            p2s[wave][lid] = x0 * x0 + x1 * x1 + x2 * x2;
            kss[wave][lid] = ks;
            if (ks >= 0) {
                float dx = x0 - mzL[ks * 3 + 0];
                float dy = x1 - mzL[ks * 3 + 1];
                float dz = x2 - mzL[ks * 3 + 2];
                acc += (huber1(dx) + huber1(dy) + huber1(dz)) * iscL[ks];
            }
        }
        // Same-wave DS ops are in-order (ISA 7.3) -> safe to read scratch written above.
        int mb = hi ? 8 : 0;
        v8f cz;                            // C input = |P|^2, matches C/D VGPR layout
        int kb[8];                         // pre-clamped dwT row base per pixel
#pragma unroll
        for (int r = 0; r < 8; ++r) {
            cz[r] = p2s[wave][mb + r];
            int km = kss[wave][mb + r];
            unsigned ukm = (unsigned)km;   // -1 -> huge -> clamps to zero row KK
            int kmc = ukm > (unsigned)KK ? KK : (int)ukm;
            kb[r] = kmc * KK + lid;        // + t*16 folds into ds_load immediate offset
        }

#pragma unroll
        for (int t = 0; t < 4; ++t) {
            // D[m][n] = 1 + ||P_m - M_n||^2 (C and the |M|^2+1 row do all the adds)
            v8f d = __builtin_amdgcn_wmma_f32_16x16x4_f32(
                false, a, false, bmt[t], (short)0, cz, false, false);
            float accT = 0.0f;
#pragma unroll
            for (int r = 0; r < 8; ++r) {
                float wv = dwT[kb[r] + t * 16];            // 0 on diag / invalid pixel
                accT = fmaf(wv, __builtin_amdgcn_rcpf(d[r]), accT);
            }
            acc = fmaf(cfR[t], accT, acc);                 // coef folds 3000/((HW-c)*sqrt(c))
        }
    }

    red[tid] = acc;
    __syncthreads();
    if (tid == 0) {
        float s = 0.0f;
        for (int i = 0; i < TPB; ++i) s += red[i];
        atomicAdd(&accum[b], s);
    }
}

__global__ void k_final(const float* __restrict__ wsc, float* __restrict__ out) {
    if (threadIdx.x == 0 && blockIdx.x == 0) {
        float s = 0.0f;
        for (int b = 0; b < BATCH; ++b)
            s += (wsc[WS_ACC + b] + wsc[WS_MSEP + b]) / wsc[WS_CT + b];
        out[0] = s / (float)BATCH;
    }
}

extern "C" void kernel_launch(void* const* d_in, const int* in_sizes, int n_in,
                              void* d_out, int out_size, void* d_ws, size_t ws_size,
                              hipStream_t stream) {
    const float* pred = (const float*)d_in[0];          // [B,3,H,W] f32
    const float* targ = (const float*)d_in[1];          // [B,3,H,W] f32
    const unsigned char* nobg = (const unsigned char*)d_in[2];  // [B] bool (1 byte each)
    const float* dw = (const float*)d_in[3];            // [K,K] f32
    const int* pal = (const int*)d_in[4];               // [K] i32
    float* ws = (float*)d_ws;
    float* out = (float*)d_out;

    k_zero<<<(WS_TOTAL + 255) / 256, 256, 0, stream>>>(ws);
    k_count<<<BATCH * CNT_BLOCKS_PER_B, TPB, 0, stream>>>(pred, targ, pal, ws);
    k_prep<<<1, BK, 0, stream>>>(nobg, dw, pal, ws);
    k_main<<<BATCH * BLOCKS_PER_B, TPB, 0, stream>>>(pred, targ, dw, pal, ws, ws + WS_ACC);
    k_final<<<1, 32, 0, stream>>>(ws, out);
    (void)in_sizes; (void)n_in; (void)out_size; (void)ws_size;
}